// CoordinateDescentRouter_46574625358485
// MI455X (gfx1250) — compile-verified
//
#include <hip/hip_runtime.h>

// ---------------------------------------------------------------------------
// CoordinateDescentRouter for MI455X (gfx1250)
//
// Shapes (from reference): x [B=16, N=8192, D=512] f32, routing_token [1,512],
// num_tokens = 1024, eps = 1, effective_k = min(1024*9/8, 8192) = 1152.
//
// Roofline: phase 1 (GEMV) reads 256 MB @ 23.3 TB/s ~= 11 us and dominates;
// phases 2/3 operate on 512 KB of scores entirely in registers/LDS.
// Phase 1 uses V_WMMA_F32_16X16X4_F32 (fp32-faithful; op is bandwidth-bound
// so the low-rate f32 WMMA path is free).
// ---------------------------------------------------------------------------

typedef float v2f __attribute__((ext_vector_type(2)));
typedef float v8f __attribute__((ext_vector_type(8)));

#define BATCH   16
#define NTOK    8192
#define DIM     512
#define T2      1024          // threads in phase-2 block
#define EPT     (NTOK / T2)   // 8 elements per thread

// ---------------------------------------------------------------------------
// Kernel 1: s[row] = dot(x[row, :], t[:])  for row in [0, BATCH*NTOK)
// One wave per 16-token tile. A tile (16x4 fp32 of x):
//   lanes 0-15 hold M=0..15, vgpr j holds K = 2*(lane>>4) + j  (ISA 7.12.2).
// B tile (4x16): we replicate t[k] across all 16 columns, using the same
//   {vgpr, lane-half} -> K mapping, so every column of D equals the dot.
// Lanes 0 and 16 (N==0 columns of the two M-halves) write D out.
// ---------------------------------------------------------------------------
__global__ void __launch_bounds__(128)
cdr_gemv_wmma_kernel(const float* __restrict__ x,
                     const float* __restrict__ t,
                     float* __restrict__ s_out) {
  __shared__ float tl[DIM];
  for (int i = threadIdx.x; i < DIM; i += 128) tl[i] = t[i];
  __syncthreads();

  const int wave = threadIdx.x >> 5;
  const int lane = threadIdx.x & 31;
  const int m    = lane & 15;     // row within tile
  const int kh   = lane >> 4;     // K half: lanes 0-15 -> K{0,1}, 16-31 -> K{2,3}

  const int tile   = blockIdx.x * 4 + wave;     // 8192 tiles total
  const int token0 = tile * 16;
  const float* xrow = x + (size_t)(token0 + m) * DIM;

  v8f c = {};
  #pragma unroll 4
  for (int kk = 0; kk < DIM; kk += 4) {
    const v2f a2 = *(const v2f*)(xrow + kk + 2 * kh);        // global b64 load
    const v2f b2 = *(const v2f*)(&tl[kk + 2 * kh]);          // ds b64 load
    // 8 args: (neg_a, A, neg_b, B, c_mod, C, reuse_a, reuse_b)
    c = __builtin_amdgcn_wmma_f32_16x16x4_f32(
        /*neg_a=*/false, a2, /*neg_b=*/false, b2,
        /*c_mod=*/(short)0, c, /*reuse_a=*/false, /*reuse_b=*/false);
  }

  // C/D layout: vgpr v, lanes 0-15 -> M=v, lanes 16-31 -> M=v+8 (N = lane%16).
  // Every N column is identical; lanes with N==0 store their 8 rows.
  if ((lane & 15) == 0) {
    float* sp = s_out + token0 + (lane >> 4) * 8;
    #pragma unroll
    for (int v = 0; v < 8; ++v) sp[v] = c[v];
  }
}

// ---------------------------------------------------------------------------
// Kernel 2: per-row coordinate descent (50 iters, eps=1) + stable top-K.
// One 1024-thread workgroup per batch row; s,b in registers (8/thread).
// LDS: 8192 x u64 sort keys (64 KB); reduction scratch aliased into keys.
// ---------------------------------------------------------------------------
__device__ __forceinline__ float wave_max(float v) {
  #pragma unroll
  for (int off = 16; off > 0; off >>= 1) v = fmaxf(v, __shfl_xor(v, off, 32));
  return v;
}
__device__ __forceinline__ float wave_sum(float v) {
  #pragma unroll
  for (int off = 16; off > 0; off >>= 1) v += __shfl_xor(v, off, 32);
  return v;
}

__device__ __forceinline__ float block_reduce(float v, float* red, int tid, bool is_max) {
  v = is_max ? wave_max(v) : wave_sum(v);
  if ((tid & 31) == 0) red[tid >> 5] = v;
  __syncthreads();
  if (tid < 32) {
    float w = red[tid];
    w = is_max ? wave_max(w) : wave_sum(w);
    if (tid == 0) red[0] = w;
  }
  __syncthreads();
  float r = red[0];
  __syncthreads();   // red is reused immediately afterwards
  return r;
}

__global__ void __launch_bounds__(T2)
cdr_descent_topk_kernel(const float* __restrict__ s_in,
                        float* __restrict__ out,   // [scores BATCH*K][indices BATCH*K]
                        int K) {
  __shared__ unsigned long long keys[NTOK];        // 64 KB
  float* red = (float*)keys;                       // alias: scratch used pre-sort only

  const int row = blockIdx.x;
  const int tid = threadIdx.x;

  float s_loc[EPT], b_loc[EPT];
  #pragma unroll
  for (int j = 0; j < EPT; ++j) {
    s_loc[j] = s_in[row * NTOK + tid + j * T2];
    b_loc[j] = -s_loc[j];                          // b0 = -s
  }

  const float logk = logf(fminf((float)K * (9.0f / 8.0f), (float)NTOK));  // log 1152

  float a = 0.0f;
  for (int it = 0; it < 50; ++it) {
    // a = logk - logsumexp(s + b)   (eps = 1)
    float mx = -3.402823466e+38f;
    float v[EPT];
    #pragma unroll
    for (int j = 0; j < EPT; ++j) { v[j] = s_loc[j] + b_loc[j]; mx = fmaxf(mx, v[j]); }
    const float M = block_reduce(mx, red, tid, /*max*/true);
    float se = 0.0f;
    #pragma unroll
    for (int j = 0; j < EPT; ++j) se += expf(v[j] - M);
    const float S = block_reduce(se, red, tid, /*max*/false);
    a = logk - (M + logf(S));
    // b = -relu(s + a)
    #pragma unroll
    for (int j = 0; j < EPT; ++j) b_loc[j] = -fmaxf(s_loc[j] + a, 0.0f);
  }

  __syncthreads();  // done with red scratch; keys region now owned by sort

  // scores = exp(min(s + a, 0)) in (0,1]; key = (~bits(score) << 32) | idx
  // ascending sort => score descending, ties broken by lower index (== lax.top_k)
  #pragma unroll
  for (int j = 0; j < EPT; ++j) {
    const int idx = tid + j * T2;
    const float sc = expf(fminf(s_loc[j] + a, 0.0f));
    const unsigned int sb = ~__float_as_uint(sc);
    keys[idx] = ((unsigned long long)sb << 32) | (unsigned int)idx;
  }
  __syncthreads();

  // In-LDS bitonic sort, N=8192, ascending.
  for (unsigned int k2 = 2; k2 <= NTOK; k2 <<= 1) {
    for (unsigned int jj = k2 >> 1; jj >= 1; jj >>= 1) {
      #pragma unroll 2
      for (unsigned int i = tid; i < NTOK; i += T2) {
        const unsigned int ixj = i ^ jj;
        if (ixj > i) {
          const bool up = ((i & k2) == 0);
          const unsigned long long va = keys[i], vb = keys[ixj];
          if ((va > vb) == up) { keys[i] = vb; keys[ixj] = va; }
        }
      }
      __syncthreads();
    }
  }

  // Outputs: forward scores are straight-through 1.0; indices as float (exact).
  for (int i = tid; i < K; i += T2) {
    out[row * K + i] = 1.0f;
    out[BATCH * K + row * K + i] =
        (float)(unsigned int)(keys[i] & 0xFFFFFFFFull);
  }
}

// ---------------------------------------------------------------------------
// Host launcher
// ---------------------------------------------------------------------------
extern "C" void kernel_launch(void* const* d_in, const int* in_sizes, int n_in,
                              void* d_out, int out_size, void* d_ws, size_t ws_size,
                              hipStream_t stream) {
  const float* x   = (const float*)d_in[0];   // [16, 8192, 512] f32
  const float* t   = (const float*)d_in[1];   // [1, 512] f32
  (void)d_in; (void)n_in; (void)in_sizes; (void)ws_size;

  float* s_buf = (float*)d_ws;                // BATCH*NTOK f32 = 512 KB scratch
  float* out   = (float*)d_out;               // [scores 16*K][indices 16*K]
  const int K  = out_size / (2 * BATCH);      // = num_tokens (1024)

  // Phase 1: WMMA GEMV.  8192 16-token tiles, 4 waves/block.
  cdr_gemv_wmma_kernel<<<dim3((BATCH * NTOK / 16) / 4), dim3(128), 0, stream>>>(
      x, t, s_buf);

  // Phase 2: coordinate descent + stable top-K, one workgroup per batch row.
  cdr_descent_topk_kernel<<<dim3(BATCH), dim3(T2), 0, stream>>>(s_buf, out, K);
}